// GNN_17875653886652
// MI455X (gfx1250) — compile-verified
//
#include <hip/hip_runtime.h>
#include <hip/hip_bf16.h>

// ---------------------------------------------------------------------------
// GNN fused kernel for MI455X (gfx1250, wave32, WMMA).
//
// Reference collapses algebraically:
//   out = ((1 + deg) * (x @ W1 + b1)) @ W2 + b2,   deg = histogram(edge_dst)
//
// Kernels:
//   1) zero_deg   : deg[] = 0                      (N ints in d_ws)
//   2) count_deg  : atomicAdd over 1M edges
//   3) gnn_fused  : per-wave 16-node tile, f32 WMMA 16x16x4 for both GEMMs
// ---------------------------------------------------------------------------

typedef __attribute__((ext_vector_type(2))) float v2f;
typedef __attribute__((ext_vector_type(8))) float v8f;

#define GNN_N_NODES 100000
#define GNN_N_EDGES 1000000
#define GNN_D_IN    64
#define GNN_D_HID   128
#define GNN_D_OUT   64

#define WAVES_PER_BLOCK 4
#define HS_LD 132   // 128 + 4 pad: column reads hit distinct banks (132 % 64 == 4)

__global__ void gnn_zero_deg(int* __restrict__ deg) {
    int i = blockIdx.x * blockDim.x + threadIdx.x;
    if (i < GNN_N_NODES) deg[i] = 0;
}

__global__ void gnn_count_deg(const long long* __restrict__ edge_dst,
                              int* __restrict__ deg) {
    int e = blockIdx.x * blockDim.x + threadIdx.x;
    if (e < GNN_N_EDGES) {
        int d = (int)edge_dst[e];
        atomicAdd(&deg[d], 1);
    }
}

__launch_bounds__(WAVES_PER_BLOCK * 32)
__global__ void gnn_fused(const float* __restrict__ x,
                          const int*  __restrict__ deg,
                          const float* __restrict__ W1,
                          const float* __restrict__ b1,
                          const float* __restrict__ W2,
                          const float* __restrict__ b2,
                          float* __restrict__ out) {
    // Wave-private H staging: 16 rows x 128 cols (padded) of f32.
    __shared__ float Hs[WAVES_PER_BLOCK][16 * HS_LD];

    const int lane  = threadIdx.x & 31;
    const int wave  = threadIdx.x >> 5;
    const bool lo   = lane < 16;
    const int  mn   = lane & 15;        // A-frag row index / B-frag column index
    const int  koff = lo ? 0 : 2;       // K sub-offset within a 4-wide K tile
    const int  rbase = lo ? 0 : 8;      // C/D row base for this half-wave

    float* hs = &Hs[wave][0];

    const int nTiles    = GNN_N_NODES / 16;                 // 6250
    const int waveGlob  = blockIdx.x * WAVES_PER_BLOCK + wave;
    const int waveStride = gridDim.x * WAVES_PER_BLOCK;

    for (int tile = waveGlob; tile < nTiles; tile += waveStride) {
        const int m0 = tile * 16;

        // ---- Load A fragments of X tile [16 x 64]: 16 K-tiles of width 4 ----
        const float* xrow = x + (size_t)(m0 + mn) * GNN_D_IN;
        v2f ax[16];
#pragma unroll
        for (int kt = 0; kt < 16; ++kt) {
            const int k0 = kt * 4 + koff;
            ax[kt].x = xrow[k0];
            ax[kt].y = xrow[k0 + 1];
        }

        // ---- Per-lane row scales (1 + degree) for the 8 C rows it owns ----
        float scale[8];
#pragma unroll
        for (int r = 0; r < 8; ++r)
            scale[r] = 1.0f + (float)deg[m0 + rbase + r];

        // ---- GEMM1: H = X @ W1 + b1  (8 N-tiles of 16, K=64) ----
#pragma unroll
        for (int j = 0; j < 8; ++j) {
            const int n0 = j * 16;
            const float bv = b1[n0 + mn];
            v8f c = { bv, bv, bv, bv, bv, bv, bv, bv };
#pragma unroll
            for (int kt = 0; kt < 16; ++kt) {
                const int k0 = kt * 4;
                v2f bf;
                bf.x = W1[(size_t)(k0 + koff)     * GNN_D_HID + n0 + mn];
                bf.y = W1[(size_t)(k0 + koff + 1) * GNN_D_HID + n0 + mn];
                c = __builtin_amdgcn_wmma_f32_16x16x4_f32(
                        false, ax[kt], false, bf, (short)0, c, false, false);
            }
            // Scale rows by (1+deg) and stage into LDS (H-slab, padded rows).
#pragma unroll
            for (int r = 0; r < 8; ++r)
                hs[(rbase + r) * HS_LD + n0 + mn] = c[r] * scale[r];
        }

        // LDS ops are in-order within a wave; compiler inserts s_wait_dscnt
        // for the RAW dependency below.

        // ---- GEMM2: OUT = Hs @ W2 + b2  (4 N-tiles of 16, K=128) ----
#pragma unroll
        for (int j = 0; j < 4; ++j) {
            const int n0 = j * 16;
            const float bv = b2[n0 + mn];
            v8f c = { bv, bv, bv, bv, bv, bv, bv, bv };
#pragma unroll
            for (int kt = 0; kt < 32; ++kt) {
                const int k0 = kt * 4;
                v2f af, bf;
                af.x = hs[mn * HS_LD + k0 + koff];
                af.y = hs[mn * HS_LD + k0 + koff + 1];
                bf.x = W2[(size_t)(k0 + koff)     * GNN_D_OUT + n0 + mn];
                bf.y = W2[(size_t)(k0 + koff + 1) * GNN_D_OUT + n0 + mn];
                c = __builtin_amdgcn_wmma_f32_16x16x4_f32(
                        false, af, false, bf, (short)0, c, false, false);
            }
#pragma unroll
            for (int r = 0; r < 8; ++r)
                out[(size_t)(m0 + rbase + r) * GNN_D_OUT + n0 + mn] = c[r];
        }
    }
}

extern "C" void kernel_launch(void* const* d_in, const int* in_sizes, int n_in,
                              void* d_out, int out_size, void* d_ws, size_t ws_size,
                              hipStream_t stream) {
    (void)in_sizes; (void)n_in; (void)out_size; (void)ws_size;

    const float*     x    = (const float*)d_in[0];
    const long long* ei   = (const long long*)d_in[1];   // int64 [2, E]
    const float*     W1   = (const float*)d_in[2];
    const float*     b1   = (const float*)d_in[3];
    const float*     W2   = (const float*)d_in[4];
    const float*     b2   = (const float*)d_in[5];
    float*           out  = (float*)d_out;

    const long long* edge_dst = ei + GNN_N_EDGES;        // row 1 of edge_index
    int* deg = (int*)d_ws;                               // N_NODES ints

    // 1) deg = 0
    {
        const int T = 256, B = (GNN_N_NODES + T - 1) / T;
        gnn_zero_deg<<<B, T, 0, stream>>>(deg);
    }
    // 2) degree histogram over edges
    {
        const int T = 256, B = (GNN_N_EDGES + T - 1) / T;
        gnn_count_deg<<<B, T, 0, stream>>>(edge_dst, deg);
    }
    // 3) fused WMMA GEMM-scale-GEMM, one 16-node tile per wave
    {
        const int nTiles = GNN_N_NODES / 16;             // 6250
        const int T = WAVES_PER_BLOCK * 32;              // 128 threads
        const int B = (nTiles + WAVES_PER_BLOCK - 1) / WAVES_PER_BLOCK;
        gnn_fused<<<B, T, 0, stream>>>(x, deg, W1, b1, W2, b2, out);
    }
}